// ROIAlign_23931557773457
// MI455X (gfx1250) — compile-verified
//
#include <hip/hip_runtime.h>

// ROI Align for MI455X (gfx1250, wave32).
// Strategy: L2-resident gather (features = 15.6 MB << 192 MB L2), weighted
// 16-point reduction done as 4 chained V_WMMA_F32_16X16X4_F32 per bin:
//   D(16ch x 16) += A(16ch x 4corners) x B(4corners x 16, columns replicated)
// One wave = (roi, 16-channel group), loops over the 49 output bins.

typedef float v2f __attribute__((ext_vector_type(2)));
typedef float v8f __attribute__((ext_vector_type(8)));

#define RA_PH 7
#define RA_PW 7
#define RA_FH 50
#define RA_FW 76
#define RA_FC 256
#define RA_SCALE (1.0f / 16.0f)

__global__ __launch_bounds__(256) void roi_align_wmma_kernel(
    const float* __restrict__ feat,   // (B, 256, 50, 76) fp32
    const float* __restrict__ rois,   // (N, 5) fp32: bidx, x1, y1, x2, y2
    float* __restrict__ out,          // (N, 256, 7, 7) fp32
    int N)
{
    const int lane = threadIdx.x & 31;
    const int wid  = blockIdx.x * 8 + (threadIdx.x >> 5);   // global wave id
    const int n    = wid >> 4;                              // roi index
    const int cg   = wid & 15;                              // channel group of 16
    if (n >= N) return;                                     // wave-uniform guard

    const int half = lane >> 4;                 // 0: (yl,*) corners, 1: (yh,*)
    const int c    = cg * 16 + (lane & 15);     // this lane's A-matrix row (channel)

    // ROI geometry (wave-uniform; broadcast through L0)
    const float* r = rois + n * 5;
    const int   b  = (int)r[0];
    const float x1 = r[1] * RA_SCALE;
    const float y1 = r[2] * RA_SCALE;
    const float x2 = r[3] * RA_SCALE;
    const float y2 = r[4] * RA_SCALE;
    const float bin_h = fmaxf(y2 - y1, 1.0f) * (1.0f / RA_PH);
    const float bin_w = fmaxf(x2 - x1, 1.0f) * (1.0f / RA_PW);

    const float* plane = feat + (size_t)(b * RA_FC + c) * (RA_FH * RA_FW);
    float*       outb  = out + (size_t)n * RA_FC * (RA_PH * RA_PW);

    for (int bin = 0; bin < RA_PH * RA_PW; ++bin) {
        const int ph = bin / RA_PW;
        const int pw = bin % RA_PW;
        v8f acc = {};

        // 4 sampling points (sr=2 x 2); each is one K=4 WMMA over its 4 corners.
        #pragma unroll
        for (int s = 0; s < 4; ++s) {
            const int sy = s >> 1, sx = s & 1;
            const float y = y1 + ((float)ph + ((float)sy + 0.5f) * 0.5f) * bin_h;
            const float x = x1 + ((float)pw + ((float)sx + 0.5f) * 0.5f) * bin_w;
            const bool valid = (y > -1.0f) && (y < (float)RA_FH) &&
                               (x > -1.0f) && (x < (float)RA_FW);
            const float yc = fminf(fmaxf(y, 0.0f), (float)(RA_FH - 1));
            const float xc = fminf(fmaxf(x, 0.0f), (float)(RA_FW - 1));
            const int yl = (int)yc;
            const int xl = (int)xc;
            const int yh = min(yl + 1, RA_FH - 1);
            const int xh = min(xl + 1, RA_FW - 1);
            const float ly = yc - (float)yl;
            const float lx = xc - (float)xl;
            const float hy = 1.0f - ly;
            const float hx = 1.0f - lx;

            // Lane-half selects corner row: half0 -> K0,K1 = (yl,xl),(yl,xh)
            //                               half1 -> K2,K3 = (yh,xl),(yh,xh)
            const int   yr = half ? yh : yl;
            const float wy = half ? ly : hy;
            const float ws = valid ? 0.25f * wy : 0.0f;  // 0.25 = sample mean

            v2f a, w;
            a.x = plane[yr * RA_FW + xl];
            a.y = plane[yr * RA_FW + xh];
            w.x = ws * hx;   // weight for left  corner (all 16 B-columns equal)
            w.y = ws * lx;   // weight for right corner

            // D = A(16x4) * B(4x16) + C  -> v_wmma_f32_16x16x4_f32
            acc = __builtin_amdgcn_wmma_f32_16x16x4_f32(
                      false, a, false, w, (short)0, acc, false, false);
        }

        // D columns are identical; VGPR j of lane-half h holds channel
        // cg*16 + 8*h + j. Lanes 0 and 16 emit the 16 channel results.
        if ((lane & 15) == 0) {
            const int cb = cg * 16 + half * 8;
            #pragma unroll
            for (int j = 0; j < 8; ++j)
                outb[(size_t)(cb + j) * (RA_PH * RA_PW) + bin] = acc[j];
        }
    }
}

extern "C" void kernel_launch(void* const* d_in, const int* in_sizes, int n_in,
                              void* d_out, int out_size, void* d_ws, size_t ws_size,
                              hipStream_t stream) {
    const float* feat = (const float*)d_in[0];   // 4*256*50*76 fp32
    const float* rois = (const float*)d_in[1];   // N*5 fp32
    float* out = (float*)d_out;                  // N*256*7*7 fp32

    const int N = in_sizes[1] / 5;               // 1024
    const int waves  = N * (RA_FC / 16);         // one wave per (roi, 16-ch group)
    const int blocks = (waves + 7) / 8;          // 8 waves (256 threads) per block

    hipLaunchKernelGGL(roi_align_wmma_kernel, dim3(blocks), dim3(256), 0, stream,
                       feat, rois, out, N);
}